// QA_CNN_54116587929806
// MI455X (gfx1250) — compile-verified
//
#include <hip/hip_runtime.h>

// ---- problem constants (from reference) ----
#define Bn 64
#define Tn 1024
#define En 128
#define Cn 5
#define Fn 64
#define Hn 2            // C/2
#define Kn (Cn * En)    // 640

// ---- tiling ----
#define TN    64                  // t-tile width per workgroup (4 n-subtiles of 16)
#define NROW  (TN + Cn - 1)       // 68 staged x rows (includes halo)
#define LDSTR 132                 // padded LDS row stride (132 % 64 = 4 banks -> no conflicts)

typedef __attribute__((ext_vector_type(2))) float v2f;
typedef __attribute__((ext_vector_type(8))) float v8f;

__global__ __launch_bounds__(128)
void qa_cnn_wmma_f32(const float* __restrict__ x,
                     const float* __restrict__ W,
                     const float* __restrict__ bias,
                     float* __restrict__ out) {
    __shared__ float xs[NROW * LDSTR];   // 68*132*4 = 35,904 B

    const int tid  = threadIdx.x;
    const int bIdx = blockIdx.x >> 4;          // T/TN = 16 t-tiles per batch
    const int t0   = (blockIdx.x & 15) * TN;

    // ---- stage x window [t0-H, t0+TN+H) into LDS, zero-padded at edges ----
    const float* xb = x + (size_t)bIdx * Tn * En;
    for (int i = tid; i < NROW * (En / 4); i += 128) {   // 2176 float4 chunks
        const int row = i >> 5;                 // En/4 = 32 chunks per row
        const int e4  = (i & 31) << 2;
        const int t   = t0 + row - Hn;
        float4 v = make_float4(0.f, 0.f, 0.f, 0.f);
        if (t >= 0 && t < Tn)
            v = *(const float4*)(xb + (size_t)t * En + e4);
        *(float4*)(&xs[row * LDSTR + e4]) = v;
    }
    __syncthreads();

    // ---- per-wave GEMM: 16 F-rows x 64 t-cols, K = 640 via 16x16x4 f32 WMMA ----
    const int lane  = tid & 31;
    const int laneM = lane & 15;     // A: M row / B,C,D: N col
    const int g     = lane >> 4;     // half-wave group
    const int f0    = (tid >> 5) << 4;

    v8f acc0 = {}, acc1 = {}, acc2 = {}, acc3 = {};
    const float* wrow = W + (size_t)(f0 + laneM) * Kn;

    #pragma unroll 4
    for (int k0 = 0; k0 < Kn; k0 += 4) {
        // A fragment (16x4): lane L holds M=L&15, K = k0 + 2*(L>>4) + {0,1}
        const v2f a = *(const v2f*)(wrow + k0 + 2 * g);

        // k-step never crosses a tap boundary: c, e0 constant within step
        const int c  = k0 >> 7;      // tap index  = k0 / E
        const int e0 = k0 & 127;     // embed col  = k0 % E
        const float* bsrc = &xs[(laneM + c) * LDSTR + e0 + g];

        // B fragment (4x16): lane L holds N=L&15; vgpr0 -> K=k0+g, vgpr1 -> K=k0+2+g
        {
            const v2f bf = { bsrc[0 * 16 * LDSTR], bsrc[0 * 16 * LDSTR + 2] };
            acc0 = __builtin_amdgcn_wmma_f32_16x16x4_f32(false, a, false, bf,
                                                         (short)0, acc0, false, false);
        }
        {
            const v2f bf = { bsrc[1 * 16 * LDSTR], bsrc[1 * 16 * LDSTR + 2] };
            acc1 = __builtin_amdgcn_wmma_f32_16x16x4_f32(false, a, false, bf,
                                                         (short)0, acc1, false, false);
        }
        {
            const v2f bf = { bsrc[2 * 16 * LDSTR], bsrc[2 * 16 * LDSTR + 2] };
            acc2 = __builtin_amdgcn_wmma_f32_16x16x4_f32(false, a, false, bf,
                                                         (short)0, acc2, false, false);
        }
        {
            const v2f bf = { bsrc[3 * 16 * LDSTR], bsrc[3 * 16 * LDSTR + 2] };
            acc3 = __builtin_amdgcn_wmma_f32_16x16x4_f32(false, a, false, bf,
                                                         (short)0, acc3, false, false);
        }
    }

    // ---- store D + bias: C/D layout -> VGPR r holds M = r + 8*g, N = laneM ----
    float* ob = out + (size_t)bIdx * Fn * Tn;
    #pragma unroll
    for (int r = 0; r < 8; ++r) {
        const int   f  = f0 + r + 8 * g;
        const float bv = bias[f];
        float* orow = ob + (size_t)f * Tn + t0 + laneM;
        orow[0]  = acc0[r] + bv;
        orow[16] = acc1[r] + bv;
        orow[32] = acc2[r] + bv;
        orow[48] = acc3[r] + bv;
    }
}

extern "C" void kernel_launch(void* const* d_in, const int* in_sizes, int n_in,
                              void* d_out, int out_size, void* d_ws, size_t ws_size,
                              hipStream_t stream) {
    (void)in_sizes; (void)n_in; (void)out_size; (void)d_ws; (void)ws_size;
    const float* x    = (const float*)d_in[0];
    const float* W    = (const float*)d_in[1];
    const float* bias = (const float*)d_in[2];
    float*       out  = (float*)d_out;

    dim3 grid(Bn * (Tn / TN));   // 64 batches * 16 t-tiles = 1024 workgroups
    dim3 block(128);             // 4 waves: one 16-row F-tile each
    qa_cnn_wmma_f32<<<grid, block, 0, stream>>>(x, W, bias, out);
}